// MambaModule_11836929868492
// MI455X (gfx1250) — compile-verified
//
#include <hip/hip_runtime.h>
#include <hip/hip_bf16.h>
#include <math.h>

#define B_SZ     8
#define SEQ_L    4096
#define DMODEL   64
#define DINNER   128
#define DSTATE   16
#define DTRANK   4
#define CONVK    4
#define NROWS    (B_SZ * SEQ_L)   // 32768 token rows
#define XDB_LD   48               // 36 (dt_low+B+C) padded to 3 WMMA tiles
#define NTW      4                // N-tiles per wave in the WMMA GEMM

typedef float v2f __attribute__((ext_vector_type(2)));
typedef float v8f __attribute__((ext_vector_type(8)));

__device__ __forceinline__ float silu_f(float v)     { return v / (1.0f + __expf(-v)); }
__device__ __forceinline__ float softplus_f(float v) { return (v > 20.0f) ? v : log1pf(__expf(v)); }

// ---------------------------------------------------------------------------
// LayerNorm: one wave32 per row of 64, lane handles elems lane and lane+32.
// ---------------------------------------------------------------------------
__global__ void ln_kernel(const float* __restrict__ x, const float* __restrict__ g,
                          const float* __restrict__ b, float* __restrict__ xn) {
    int wave = (blockIdx.x * blockDim.x + threadIdx.x) >> 5;
    int lane = threadIdx.x & 31;
    if (wave >= NROWS) return;
    const float* row = x + (size_t)wave * DMODEL;
    float v0 = row[lane], v1 = row[lane + 32];
    float s = v0 + v1;
    #pragma unroll
    for (int m = 16; m; m >>= 1) s += __shfl_xor(s, m, 32);
    float mean = s * (1.0f / 64.0f);
    float d0 = v0 - mean, d1 = v1 - mean;
    float q = d0 * d0 + d1 * d1;
    #pragma unroll
    for (int m = 16; m; m >>= 1) q += __shfl_xor(q, m, 32);
    float rstd = rsqrtf(q * (1.0f / 64.0f) + 1e-5f);
    float* out = xn + (size_t)wave * DMODEL;
    out[lane]      = d0 * rstd * g[lane]      + b[lane];
    out[lane + 32] = d1 * rstd * g[lane + 32] + b[lane + 32];
}

// ---------------------------------------------------------------------------
// fp32 WMMA GEMM with 4 N-tiles per wave:  Y[M,N] = X[M,K]*W[N,K]^T (+Y if accum)
// One A-fragment load feeds 4 v_wmma_f32_16x16x4_f32 per K-step (A reuse).
// ntiles = number of stored (padded-buffer) tiles; groups = ceil(ntiles/NTW).
// Weight rows beyond Nact clamped+zeroed (per-lane select); store guards are
// wave-uniform, so EXEC is all-ones across every WMMA as required.
// ---------------------------------------------------------------------------
__global__ void gemm_wmma_f32(const float* __restrict__ X, int ldx,
                              const float* __restrict__ W, int ldw, int Nact,
                              float* __restrict__ Y, int ldy,
                              int mtiles, int ntiles, int groups,
                              int kdim, int accum) {
    int wid = (blockIdx.x * blockDim.x + threadIdx.x) >> 5;
    if (wid >= mtiles * groups) return;          // wave-uniform guard
    int lane = threadIdx.x & 31;
    int hi   = lane >> 4;                        // 0: K pair {0,1}, 1: {2,3}
    int l16  = lane & 15;
    int mt     = wid / groups;
    int ntbase = (wid % groups) * NTW;

    int mrow = mt * 16 + l16;                    // A rows 0..15 in both halves
    const float* xr = X + (size_t)mrow * ldx + hi * 2;

    int          ncol[NTW];
    bool         wv[NTW];
    const float* wr[NTW];
    #pragma unroll
    for (int j = 0; j < NTW; ++j) {
        int nt  = ntbase + j;
        ncol[j] = nt * 16 + l16;
        wv[j]   = (nt < ntiles) && (ncol[j] < Nact);
        wr[j]   = W + (size_t)(wv[j] ? ncol[j] : 0) * ldw + hi * 2;
    }

    v8f c[NTW];
    #pragma unroll
    for (int j = 0; j < NTW; ++j) {
        c[j] = (v8f){};
        if (accum && (ntbase + j) < ntiles) {
            #pragma unroll
            for (int r = 0; r < 8; ++r)
                c[j][r] = Y[(size_t)(mt * 16 + hi * 8 + r) * ldy + ncol[j]];
        }
    }

    for (int k = 0; k < kdim; k += 4) {
        v2f a = *(const v2f*)(xr + k);           // loaded once, used 4x
        #pragma unroll
        for (int j = 0; j < NTW; ++j) {
            v2f bb = *(const v2f*)(wr[j] + k);
            if (!wv[j]) { bb[0] = 0.0f; bb[1] = 0.0f; }
            c[j] = __builtin_amdgcn_wmma_f32_16x16x4_f32(
                       false, a, false, bb, (short)0, c[j], false, false);
        }
    }

    #pragma unroll
    for (int j = 0; j < NTW; ++j) {
        if ((ntbase + j) < ntiles) {             // wave-uniform store guard
            #pragma unroll
            for (int r = 0; r < 8; ++r)
                Y[(size_t)(mt * 16 + hi * 8 + r) * ldy + ncol[j]] = c[j][r];
        }
    }
}

// ---------------------------------------------------------------------------
// Causal (rev=0) / anti-causal (rev=1) depthwise conv K=4 + bias + SiLU.
// xi lives in xz cols [0,128); result -> xc[r*128+ch].
// ---------------------------------------------------------------------------
__global__ void conv_silu_kernel(const float* __restrict__ xz, const float* __restrict__ cw,
                                 const float* __restrict__ cb, float* __restrict__ xc, int rev) {
    int idx = blockIdx.x * blockDim.x + threadIdx.x;
    if (idx >= NROWS * DINNER) return;
    int ch = idx & (DINNER - 1);
    int r  = idx >> 7;
    int l  = r & (SEQ_L - 1);
    int base = r - l;                            // row index of l==0 in this batch
    float acc = cb[ch];
    #pragma unroll
    for (int j = 0; j < CONVK; ++j) {
        int pos = rev ? (l + (CONVK - 1) - j) : (l - (CONVK - 1) + j);
        if (pos >= 0 && pos < SEQ_L)
            acc += cw[ch * CONVK + j] * xz[(size_t)(base + pos) * (2 * DINNER) + ch];
    }
    xc[idx] = silu_f(acc);
}

// dt = softplus(dt_raw + bias)
__global__ void dt_softplus_kernel(float* __restrict__ dtb, const float* __restrict__ bias) {
    int idx = blockIdx.x * blockDim.x + threadIdx.x;
    if (idx >= NROWS * DINNER) return;
    dtb[idx] = softplus_f(dtb[idx] + bias[idx & (DINNER - 1)]);
}

// ---------------------------------------------------------------------------
// Selective scan: each SSM state n gets its own lane (16 lanes per (b,ch)).
// Serial over L; y = sum_n h[n]*C[n] via width-16 shfl_xor reduction.
// 2-deep unroll + global_prefetch_b8 let next-step loads hide behind the
// v_exp_f32 + FMA dependency chain.
// ---------------------------------------------------------------------------
__global__ void scan_kernel(const float* __restrict__ xc, const float* __restrict__ dtb,
                            const float* __restrict__ xdb, const float* __restrict__ A_log,
                            float* __restrict__ ys, int rev) {
    int g    = blockIdx.x * blockDim.x + threadIdx.x;   // 16384 threads
    int n    = g & (DSTATE - 1);
    int pair = g >> 4;                                   // (b,ch): 0..1023
    if (pair >= B_SZ * DINNER) return;
    int b  = pair >> 7;
    int ch = pair & (DINNER - 1);

    float A = -__expf(A_log[ch * DSTATE + n]);
    float h = 0.0f;
    const size_t rbase = (size_t)b * SEQ_L;

    #pragma unroll 2
    for (int t = 0; t < SEQ_L; ++t) {
        int l = rev ? (SEQ_L - 1 - t) : t;
        size_t r = rbase + l;
        float dtv = dtb[r * DINNER + ch];
        float xv  = xc [r * DINNER + ch];
        const float* row = xdb + r * XDB_LD;
        float Bv = row[DTRANK + n];
        float Cv = row[DTRANK + DSTATE + n];

        int ln = rev ? (l - 2) : (l + 2);
        if (ln >= 0 && ln < SEQ_L)
            __builtin_prefetch(xdb + (rbase + ln) * XDB_LD, 0, 0);

        float dA = __expf(dtv * A);
        h = dA * h + (dtv * xv) * Bv;
        float y = h * Cv;
        #pragma unroll
        for (int m = 8; m; m >>= 1) y += __shfl_xor(y, m, 16);
        if (n == 0) ys[r * DINNER + ch] = y;
    }
}

// y = (ys + xc*D) * silu(z)   with z = xz cols [128,256)
__global__ void gate_kernel(const float* __restrict__ ys, const float* __restrict__ xc,
                            const float* __restrict__ xz, const float* __restrict__ Dp,
                            float* __restrict__ yg) {
    int idx = blockIdx.x * blockDim.x + threadIdx.x;
    if (idx >= NROWS * DINNER) return;
    int ch = idx & (DINNER - 1);
    int r  = idx >> 7;
    float z = xz[(size_t)r * (2 * DINNER) + DINNER + ch];
    yg[idx] = (ys[idx] + xc[idx] * Dp[ch]) * silu_f(z);
}

// ---------------------------------------------------------------------------
extern "C" void kernel_launch(void* const* d_in, const int* in_sizes, int n_in,
                              void* d_out, int out_size, void* d_ws, size_t ws_size,
                              hipStream_t stream) {
    const float* x        = (const float*)d_in[0];
    const float* ln_g     = (const float*)d_in[1];
    const float* ln_b     = (const float*)d_in[2];
    const float* in_w     = (const float*)d_in[3];   // [2,256,64]
    const float* conv_w   = (const float*)d_in[4];   // [2,128,4]
    const float* conv_b   = (const float*)d_in[5];   // [2,128]
    const float* xp_w     = (const float*)d_in[6];   // [2,36,128]
    const float* dtp_w    = (const float*)d_in[7];   // [2,128,4]
    const float* dt_bias  = (const float*)d_in[8];   // [2,128]
    const float* A_log    = (const float*)d_in[9];   // [2,128,16]
    const float* Dp       = (const float*)d_in[10];  // [2,128]
    const float* out_w    = (const float*)d_in[11];  // [2,64,128]
    float* out = (float*)d_out;

    float* ws  = (float*)d_ws;
    float* xn  = ws;                                  // 32768*64
    float* xz  = xn  + (size_t)NROWS * DMODEL;        // 32768*256
    float* xc  = xz  + (size_t)NROWS * 2 * DINNER;    // 32768*128
    float* xdb = xc  + (size_t)NROWS * DINNER;        // 32768*48
    float* dtb = xdb + (size_t)NROWS * XDB_LD;        // 32768*128
    float* ysb = dtb + (size_t)NROWS * DINNER;        // 32768*128
    float* ygb = ysb + (size_t)NROWS * DINNER;        // 32768*128

    const int mtiles    = NROWS / 16;                 // 2048
    const int ew_blocks = (NROWS * DINNER) / 256;     // elementwise grids

    ln_kernel<<<NROWS / 8, 256, 0, stream>>>(x, ln_g, ln_b, xn);

    for (int d = 0; d < 2; ++d) {
        // in_proj: [32768,64] x [256,64]^T -> xz[32768,256]; 16 tiles -> 4 groups
        gemm_wmma_f32<<<(mtiles * 4) / 8, 256, 0, stream>>>(
            xn, DMODEL, in_w + (size_t)d * 2 * DINNER * DMODEL, DMODEL, 2 * DINNER,
            xz, 2 * DINNER, mtiles, 16, 4, DMODEL, 0);

        // depthwise conv + SiLU (anti-causal for d==1)
        conv_silu_kernel<<<ew_blocks, 256, 0, stream>>>(
            xz, conv_w + (size_t)d * DINNER * CONVK, conv_b + (size_t)d * DINNER, xc, d);

        // x_proj: [32768,128] x [36,128]^T -> xdb[32768,48]; 3 tiles -> 1 group
        gemm_wmma_f32<<<(mtiles * 1) / 8, 256, 0, stream>>>(
            xc, DINNER, xp_w + (size_t)d * (DTRANK + 2 * DSTATE) * DINNER, DINNER,
            DTRANK + 2 * DSTATE, xdb, XDB_LD, mtiles, 3, 1, DINNER, 0);

        // dt_proj: [32768,4] x [128,4]^T -> dtb[32768,128]; 8 tiles -> 2 groups
        gemm_wmma_f32<<<(mtiles * 2) / 8, 256, 0, stream>>>(
            xdb, XDB_LD, dtp_w + (size_t)d * DINNER * DTRANK, DTRANK, DINNER,
            dtb, DINNER, mtiles, 8, 2, DTRANK, 0);

        dt_softplus_kernel<<<ew_blocks, 256, 0, stream>>>(dtb, dt_bias + (size_t)d * DINNER);

        // selective scan (serial over L, 16 lanes per (b,ch))
        scan_kernel<<<(B_SZ * DINNER * DSTATE) / 256, 256, 0, stream>>>(
            xc, dtb, xdb, A_log + (size_t)d * DINNER * DSTATE, ysb, d);

        gate_kernel<<<ew_blocks, 256, 0, stream>>>(ysb, xc, xz, Dp + (size_t)d * DINNER, ygb);

        // out_proj: [32768,128] x [64,128]^T -> d_out; 4 tiles -> 1 group (accum d==1)
        gemm_wmma_f32<<<(mtiles * 1) / 8, 256, 0, stream>>>(
            ygb, DINNER, out_w + (size_t)d * DMODEL * DINNER, DINNER, DMODEL,
            out, DMODEL, mtiles, 4, 1, DINNER, d);
    }
}